// MultiHeadAttention_62740882259973
// MI455X (gfx1250) — compile-verified
//
#include <hip/hip_runtime.h>
#include <math.h>

#define B_  2
#define S_  2048
#define D_  768
#define H_  12
#define DK_ 64
#define EPS_ 1e-5f

#define TM 64
#define TN 64
#define KC 16
#define KP (KC + 2)   /* gemm LDS row pad: 18 floats -> 8B-aligned pairs */
#define TP (TN + 4)   /* 64x64 tile LDS row pad: 68 floats -> 16B aligned, 4r bank spread */
#define NT 256

typedef float v2f __attribute__((ext_vector_type(2)));
typedef float v8f __attribute__((ext_vector_type(8)));

__device__ __forceinline__ v8f wmma_f32(v2f a, v2f b, v8f c) {
  // D = A(16x4) x B(4x16) + C(16x16), all fp32, wave32
  return __builtin_amdgcn_wmma_f32_16x16x4_f32(false, a, false, b, (short)0, c,
                                               false, false);
}

// ---------------------------------------------------------------------------
// Y = X * W^T + bias   (X: [M x K] row-major, W: [N x K] row-major)
// MODE 0: scatter into head-major ws buffer [B,H,S,DK]
// MODE 1: += residual, store plain [M x N] (output projection)
// M = B*S = 4096, N = D = 768, K = D = 768
// ---------------------------------------------------------------------------
template <int MODE>
__global__ __launch_bounds__(NT) void gemm_xwt_kernel(
    const float* __restrict__ X, const float* __restrict__ W,
    const float* __restrict__ bias, const float* __restrict__ resid,
    float* __restrict__ Y) {
  constexpr int K = D_;
  __shared__ float Al[TM][KP];
  __shared__ float Bl[TN][KP];
  const int tid  = threadIdx.x;
  const int lane = tid & 31, wave = tid >> 5;
  const int half = lane >> 4, lr16 = lane & 15;
  const int wm = wave >> 1, wn = wave & 1;           // 4 x 2 wave grid
  const int mBase = blockIdx.y * TM;
  const int nBase = blockIdx.x * TN;
  const int lr = tid >> 2;                           // 0..63 tile row (loads)
  const int lc = (tid & 3) * 4;                      // 0,4,8,12

  v8f acc0 = {}; v8f acc1 = {};
  for (int kb = 0; kb < K; kb += KC) {
    float4 a4 = *(const float4*)(X + (size_t)(mBase + lr) * K + kb + lc);
    float4 b4 = *(const float4*)(W + (size_t)(nBase + lr) * K + kb + lc);
    Al[lr][lc + 0] = a4.x; Al[lr][lc + 1] = a4.y;
    Al[lr][lc + 2] = a4.z; Al[lr][lc + 3] = a4.w;
    Bl[lr][lc + 0] = b4.x; Bl[lr][lc + 1] = b4.y;
    Bl[lr][lc + 2] = b4.z; Bl[lr][lc + 3] = b4.w;
    if (kb + KC < K) {
      __builtin_prefetch(X + (size_t)(mBase + lr) * K + kb + KC + lc, 0, 1);
      __builtin_prefetch(W + (size_t)(nBase + lr) * K + kb + KC + lc, 0, 1);
    }
    __syncthreads();
#pragma unroll
    for (int k0 = 0; k0 < KC; k0 += 4) {
      const int ka = k0 + 2 * half;
      v2f a, b0, b1;
      a.x  = Al[wm * 16 + lr16][ka];      a.y  = Al[wm * 16 + lr16][ka + 1];
      b0.x = Bl[wn * 32 + lr16][ka];      b0.y = Bl[wn * 32 + lr16][ka + 1];
      b1.x = Bl[wn * 32 + 16 + lr16][ka]; b1.y = Bl[wn * 32 + 16 + lr16][ka + 1];
      acc0 = wmma_f32(a, b0, acc0);
      acc1 = wmma_f32(a, b1, acc1);
    }
    __syncthreads();
  }

  const int rowTop = mBase + wm * 16 + 8 * half;
  const int col0 = nBase + wn * 32 + lr16;
  const int col1 = col0 + 16;
#pragma unroll
  for (int i = 0; i < 8; ++i) {
    const int m = rowTop + i;
    float v0 = acc0[i] + bias[col0];
    float v1 = acc1[i] + bias[col1];
    if (MODE == 1) {
      v0 += resid[(size_t)m * D_ + col0];
      v1 += resid[(size_t)m * D_ + col1];
      Y[(size_t)m * D_ + col0] = v0;
      Y[(size_t)m * D_ + col1] = v1;
    } else {
      const int b = m >> 11, s = m & (S_ - 1);
      Y[(((size_t)b * H_ + (col0 >> 6)) * S_ + s) * DK_ + (col0 & 63)] = v0;
      Y[(((size_t)b * H_ + (col1 >> 6)) * S_ + s) * DK_ + (col1 & 63)] = v1;
    }
  }
}

// ---------------------------------------------------------------------------
// Fused attention: per (z = b*H+h, 64-row q block)
//   phase 1: S = QK^T/8 (WMMA), causal mask, raw tiles -> attn (L2-resident),
//            online row max m and rescaled sumexp l in LDS
//   phase 2: re-read raw tiles (L2), p = exp(s-m)/l, overwrite attn with p,
//            O += p @ V (WMMA), scatter O into concat-head layout
//   tail   : zero-fill strictly-masked tiles of attn
// ---------------------------------------------------------------------------
__global__ __launch_bounds__(NT) void fused_attn_kernel(
    const float* __restrict__ Qh, const float* __restrict__ Kh,
    const float* __restrict__ Vh, float* __restrict__ attn,
    float* __restrict__ Hc) {
  __shared__ float Qt[TM][TP];      // Q tile   [q][d]
  __shared__ float KVt[TM][TP];     // K tile [k][d] (ph1) / V tile [k][d] (ph2)
  __shared__ float St[TM][TP];      // score / p tile [q][k]
  __shared__ float sm_m[TM];        // running row max
  __shared__ float sm_l[TM];        // running row sumexp
  __shared__ float sm_mnew[TM];
  __shared__ float sm_scale[TM];    // exp(m_old - m_new) | later 1/l
  __shared__ float sm_part[TM][4];

  const int tid  = threadIdx.x;
  const int lane = tid & 31, wave = tid >> 5;
  const int half = lane >> 4, lr16 = lane & 15;
  const int wm = wave >> 1, wn = wave & 1;
  const int qBase = blockIdx.x * TM;
  const int z = blockIdx.y;
  const int b = z / H_, h = z % H_;
  const float ninf = -__builtin_huge_valf();

  float* aZ = attn + (size_t)z * S_ * S_;
  const float* Qp = Qh + ((size_t)z * S_ + qBase) * DK_;
  const float* Kp = Kh + (size_t)z * S_ * DK_;
  const float* Vp = Vh + (size_t)z * S_ * DK_;

  // tile copy helper indices: 1024 float4 slots = 64 rows x 16 float4
  const int cr = tid >> 4 << 2;            // base row for this thread's 4 slots? no:
  // use flat: slot j: idx4 = tid + j*NT -> r = idx4>>4, c = (idx4&15)*4
  (void)cr;

  // load Q tile once
#pragma unroll
  for (int j = 0; j < 4; ++j) {
    const int idx4 = tid + j * NT;
    const int r = idx4 >> 4, c = (idx4 & 15) * 4;
    const float4 q4 = *(const float4*)(Qp + (size_t)r * DK_ + c);
    *(float4*)(&Qt[r][c]) = q4;
  }
  if (tid < TM) { sm_m[tid] = ninf; sm_l[tid] = 0.f; }
  __syncthreads();

  const int kmax = qBase + TM;             // causal cap (exclusive)
  const int rowTop = wm * 16 + 8 * half;   // tile-local
  const int c0 = wn * 32 + lr16, c1 = c0 + 16;
  const int pr = tid >> 2;                 // stats: row per thread group of 4
  const int pc = (tid & 3) * 16;           // stats: 16-col segment

  // ---------------- phase 1: scores + online softmax stats ----------------
  for (int kb = 0; kb < kmax; kb += TN) {
#pragma unroll
    for (int j = 0; j < 4; ++j) {
      const int idx4 = tid + j * NT;
      const int r = idx4 >> 4, c = (idx4 & 15) * 4;
      *(float4*)(&KVt[r][c]) = *(const float4*)(Kp + (size_t)(kb + r) * DK_ + c);
    }
    __syncthreads();

    v8f acc0 = {}; v8f acc1 = {};
#pragma unroll
    for (int k0 = 0; k0 < DK_; k0 += 4) {
      const int ka = k0 + 2 * half;
      v2f a, b0, b1;
      a.x  = Qt[wm * 16 + lr16][ka];        a.y  = Qt[wm * 16 + lr16][ka + 1];
      b0.x = KVt[wn * 32 + lr16][ka];       b0.y = KVt[wn * 32 + lr16][ka + 1];
      b1.x = KVt[wn * 32 + 16 + lr16][ka];  b1.y = KVt[wn * 32 + 16 + lr16][ka + 1];
      acc0 = wmma_f32(a, b0, acc0);
      acc1 = wmma_f32(a, b1, acc1);
    }
    // scale + causal mask -> St
#pragma unroll
    for (int i = 0; i < 8; ++i) {
      const int r = rowTop + i;
      const int gq = qBase + r;
      St[r][c0] = (kb + c0 <= gq) ? acc0[i] * 0.125f : ninf;
      St[r][c1] = (kb + c1 <= gq) ? acc1[i] * 0.125f : ninf;
    }
    __syncthreads();

    // stream raw tile to attn (dirty in L2; overwritten in phase 2)
#pragma unroll
    for (int j = 0; j < 4; ++j) {
      const int idx4 = tid + j * NT;
      const int r = idx4 >> 4, c = (idx4 & 15) * 4;
      *(float4*)(aZ + (size_t)(qBase + r) * S_ + kb + c) = *(const float4*)(&St[r][c]);
    }

    // row max partials
    float pm = ninf;
#pragma unroll
    for (int c = 0; c < 16; ++c) pm = fmaxf(pm, St[pr][pc + c]);
    sm_part[pr][tid & 3] = pm;
    __syncthreads();
    if (tid < TM) {
      const float tm = fmaxf(fmaxf(sm_part[tid][0], sm_part[tid][1]),
                             fmaxf(sm_part[tid][2], sm_part[tid][3]));
      const float mo = sm_m[tid];
      const float mn = fmaxf(mo, tm);
      sm_mnew[tid]  = mn;
      sm_scale[tid] = (mo == ninf) ? 0.f : __expf(mo - mn);
    }
    __syncthreads();

    // row sumexp partials
    const float mn = sm_mnew[pr];
    float ps = 0.f;
#pragma unroll
    for (int c = 0; c < 16; ++c) {
      const float s = St[pr][pc + c];
      ps += (s == ninf) ? 0.f : __expf(s - mn);
    }
    sm_part[pr][tid & 3] = ps;
    __syncthreads();
    if (tid < TM) {
      sm_l[tid] = sm_l[tid] * sm_scale[tid] +
                  (sm_part[tid][0] + sm_part[tid][1] +
                   sm_part[tid][2] + sm_part[tid][3]);
      sm_m[tid] = sm_mnew[tid];
    }
    __syncthreads();
  }

  // ---------------- phase 2: normalize + O = p @ V ----------------
  if (tid < TM) sm_scale[tid] = 1.0f / sm_l[tid];   // reuse as 1/l
  __syncthreads();

  v8f o0 = {}; v8f o1 = {};
  for (int kb = 0; kb < kmax; kb += TN) {
    // raw -> p, write final attn, stage p in St
#pragma unroll
    for (int j = 0; j < 4; ++j) {
      const int idx4 = tid + j * NT;
      const int r = idx4 >> 4, c = (idx4 & 15) * 4;
      float4 s4 = *(const float4*)(aZ + (size_t)(qBase + r) * S_ + kb + c);
      const float m = sm_m[r], inv = sm_scale[r];
      float4 p4;
      p4.x = (s4.x == ninf) ? 0.f : __expf(s4.x - m) * inv;
      p4.y = (s4.y == ninf) ? 0.f : __expf(s4.y - m) * inv;
      p4.z = (s4.z == ninf) ? 0.f : __expf(s4.z - m) * inv;
      p4.w = (s4.w == ninf) ? 0.f : __expf(s4.w - m) * inv;
      *(float4*)(aZ + (size_t)(qBase + r) * S_ + kb + c) = p4;
      *(float4*)(&St[r][c]) = p4;
    }
    // V tile [k][d]
#pragma unroll
    for (int j = 0; j < 4; ++j) {
      const int idx4 = tid + j * NT;
      const int r = idx4 >> 4, c = (idx4 & 15) * 4;
      *(float4*)(&KVt[r][c]) = *(const float4*)(Vp + (size_t)(kb + r) * DK_ + c);
    }
    __syncthreads();
#pragma unroll
    for (int k0 = 0; k0 < TN; k0 += 4) {
      const int ka = k0 + 2 * half;
      v2f a, b0, b1;
      a.x  = St[wm * 16 + lr16][ka];  a.y  = St[wm * 16 + lr16][ka + 1];
      b0.x = KVt[ka][wn * 32 + lr16];       b0.y = KVt[ka + 1][wn * 32 + lr16];
      b1.x = KVt[ka][wn * 32 + 16 + lr16];  b1.y = KVt[ka + 1][wn * 32 + 16 + lr16];
      o0 = wmma_f32(a, b0, o0);
      o1 = wmma_f32(a, b1, o1);
    }
    __syncthreads();
  }

  // O -> concat-head layout [B,S,D]
#pragma unroll
  for (int i = 0; i < 8; ++i) {
    const int s = qBase + rowTop + i;
    Hc[((size_t)b * S_ + s) * D_ + h * DK_ + c0] = o0[i];
    Hc[((size_t)b * S_ + s) * D_ + h * DK_ + c1] = o1[i];
  }

  // zero-fill fully masked tiles (softmax of -inf rows of tiles -> exact 0)
  const float4 z4 = make_float4(0.f, 0.f, 0.f, 0.f);
  for (int kb = kmax; kb < S_; kb += TN) {
#pragma unroll
    for (int j = 0; j < 4; ++j) {
      const int idx4 = tid + j * NT;
      const int r = idx4 >> 4, c = (idx4 & 15) * 4;
      *(float4*)(aZ + (size_t)(qBase + r) * S_ + kb + c) = z4;
    }
  }
}

// ---------------------------------------------------------------------------
// LayerNorm over D_ = 768, in place on Y, affine gamma/beta
// ---------------------------------------------------------------------------
__global__ __launch_bounds__(NT) void ln_kernel(
    float* __restrict__ Y, const float* __restrict__ gamma,
    const float* __restrict__ beta) {
  float* p = Y + (size_t)blockIdx.x * D_;
  __shared__ float red[NT];
  const int tid = threadIdx.x;
  float x0 = p[tid], x1 = p[tid + NT], x2 = p[tid + 2 * NT];

  red[tid] = x0 + x1 + x2; __syncthreads();
  for (int s = NT / 2; s > 0; s >>= 1) {
    if (tid < s) red[tid] += red[tid + s];
    __syncthreads();
  }
  const float mu = red[0] * (1.0f / D_); __syncthreads();

  const float d0 = x0 - mu, d1 = x1 - mu, d2 = x2 - mu;
  red[tid] = d0 * d0 + d1 * d1 + d2 * d2; __syncthreads();
  for (int s = NT / 2; s > 0; s >>= 1) {
    if (tid < s) red[tid] += red[tid + s];
    __syncthreads();
  }
  const float rs = rsqrtf(red[0] * (1.0f / D_) + EPS_);

  p[tid]          = d0 * rs * gamma[tid]          + beta[tid];
  p[tid + NT]     = d1 * rs * gamma[tid + NT]     + beta[tid + NT];
  p[tid + 2 * NT] = d2 * rs * gamma[tid + 2 * NT] + beta[tid + 2 * NT];
}

// ---------------------------------------------------------------------------
extern "C" void kernel_launch(void* const* d_in, const int* in_sizes, int n_in,
                              void* d_out, int out_size, void* d_ws,
                              size_t ws_size, hipStream_t stream) {
  const float* q     = (const float*)d_in[0];
  const float* k     = (const float*)d_in[1];
  const float* v     = (const float*)d_in[2];
  // d_in[3] = causal mask (known analytically, unused)
  const float* Wq    = (const float*)d_in[4];
  const float* bq    = (const float*)d_in[5];
  const float* Wk    = (const float*)d_in[6];
  const float* bk    = (const float*)d_in[7];
  const float* Wv    = (const float*)d_in[8];
  const float* bv    = (const float*)d_in[9];
  const float* Wo    = (const float*)d_in[10];
  const float* bo    = (const float*)d_in[11];
  const float* gamma = (const float*)d_in[12];
  const float* beta  = (const float*)d_in[13];

  float* out  = (float*)d_out;                       // [B,S,D]
  float* attn = out + (size_t)B_ * S_ * D_;          // [B,H,S,S]

  float* ws = (float*)d_ws;
  const size_t bhsd = (size_t)B_ * H_ * S_ * DK_;    // 3,145,728 floats
  float* Qh = ws;
  float* Kh = Qh + bhsd;
  float* Vh = Kh + bhsd;
  float* Hc = Vh + bhsd;                             // [B,S,D] concat heads

  dim3 blk(NT);
  dim3 gProj(D_ / TN, (B_ * S_) / TM);               // 12 x 64
  gemm_xwt_kernel<0><<<gProj, blk, 0, stream>>>(q, Wq, bq, nullptr, Qh);
  gemm_xwt_kernel<0><<<gProj, blk, 0, stream>>>(k, Wk, bk, nullptr, Kh);
  gemm_xwt_kernel<0><<<gProj, blk, 0, stream>>>(v, Wv, bv, nullptr, Vh);

  dim3 gAttn(S_ / TM, B_ * H_);                      // 32 x 24
  fused_attn_kernel<<<gAttn, blk, 0, stream>>>(Qh, Kh, Vh, attn, Hc);

  gemm_xwt_kernel<1><<<gProj, blk, 0, stream>>>(Hc, Wo, bo, q, out);

  ln_kernel<<<dim3(B_ * S_), blk, 0, stream>>>(out, gamma, beta);
}